// TripletLossAlignedReID_19361712571336
// MI455X (gfx1250) — compile-verified
//
#include <hip/hip_runtime.h>
#include <hip/hip_bf16.h>
#include <stdint.h>

typedef float v2f __attribute__((ext_vector_type(2)));
typedef float v8f __attribute__((ext_vector_type(8)));

#define N_SAMP 4096
#define DIM    2048
#define TILE   128
#define NTILE  (N_SAMP / TILE)            // 32 -> 528 lower-triangle tiles
#define KT     32
#define KPAD   (KT + 4)   // float4-aligned rows, conflict-free b64 frag loads
#define NLOC   8
#define DLOC   128
#define MARGIN_F 0.3f

// ---------------------------------------------------------------------------
// ws layout:
//   sq       : float[4096]  @ 0        (16 KB)
//   ap       : u64[4096]    @ 16384    (32 KB)  packed (dist_bits<<32)|~col  (max)
//   an       : u64[4096]    @ 49152    (32 KB)  packed (dist_bits<<32)| col  (min)
//   p_inds   : int[4096]    @ 81920
//   n_inds   : int[4096]    @ 98304
//   sums     : float[2]     @ 114688   (global-loss sum, local-loss sum)
// ---------------------------------------------------------------------------

__global__ void init_kernel(unsigned long long* __restrict__ ap,
                            unsigned long long* __restrict__ an,
                            float* __restrict__ sums) {
    int i = blockIdx.x * blockDim.x + threadIdx.x;
    if (i < N_SAMP) { ap[i] = 0ull; an[i] = ~0ull; }
    if (i < 2) sums[i] = 0.0f;
}

__global__ __launch_bounds__(256)
void sq_kernel(const float* __restrict__ X, float* __restrict__ sq) {
    const int row = blockIdx.x;
    const int t = threadIdx.x;
    float s = 0.0f;
#pragma unroll
    for (int j = 0; j < 8; ++j) {
        float v = X[(size_t)row * DIM + t + 256 * j];
        s += v * v;
    }
    __shared__ float red[256];
    red[t] = s; __syncthreads();
    for (int w = 128; w > 0; w >>= 1) {
        if (t < w) red[t] += red[t + w];
        __syncthreads();
    }
    if (t == 0) sq[row] = red[0];
}

// Fused fp32 WMMA GEMM (G = X X^T, lower-triangle tiles only) + distance
// epilogue + hard-example mining for BOTH the tile's rows (bm+.) and, via
// symmetry, its columns' rows (bn+.).  Block computes a 128x128 tile;
// 8 waves, each a 32x64 slab.  LDS tiles row-major so each WMMA f32
// fragment is one aligned ds_load_b64; next K-chunk prefetched in registers.
__global__ __launch_bounds__(256, 1)
void gemm_mine_kernel(const float* __restrict__ X,
                      const int*   __restrict__ targets,
                      const float* __restrict__ sq,
                      unsigned long long* __restrict__ ap,
                      unsigned long long* __restrict__ an) {
    __shared__ float As[TILE][KPAD];
    __shared__ float Bs[TILE][KPAD];
    __shared__ float sqA[TILE], sqB[TILE];
    __shared__ int   tA[TILE],  tB[TILE];
    __shared__ unsigned long long apL[TILE],  anL[TILE];   // rows bm+.
    __shared__ unsigned long long apLB[TILE], anLB[TILE];  // rows bn+.

    // Decode lower-triangle tile (bmT >= bnT) from linear block index.
    const int t0 = blockIdx.x;
    int bmT = 0;
    while ((bmT + 1) * (bmT + 2) / 2 <= t0) ++bmT;
    const int bnT = t0 - bmT * (bmT + 1) / 2;

    const int tid   = threadIdx.x;
    const int wid   = tid >> 5;
    const int lane  = tid & 31;
    const int half  = lane >> 4;
    const int l16   = lane & 15;
    const int bm    = bmT * TILE;
    const int bn    = bnT * TILE;
    const int waveM = wid >> 1;          // 0..3 -> 32-row slab
    const int waveN = wid & 1;           // 0..1 -> 64-col slab

    if (tid < TILE) {
        sqA[tid] = sq[bm + tid];
        sqB[tid] = sq[bn + tid];
        tA[tid]  = targets[bm + tid];
        tB[tid]  = targets[bn + tid];
        apL[tid]  = 0ull;  anL[tid]  = ~0ull;
        apLB[tid] = 0ull;  anLB[tid] = ~0ull;
    }

    // Per-thread staging slot: 4 float4 per matrix per chunk (256 thr cover
    // 128 rows x 32 k).
    const int s_row = tid >> 3;          // 0..31, stepped over 4 row-groups
    const int s_k4  = (tid & 7) * 4;     // 0,4,...,28

    float4 ra[4], rb[4];
#pragma unroll
    for (int i = 0; i < 4; ++i) {
        int row = s_row + 32 * i;
        ra[i] = *reinterpret_cast<const float4*>(&X[(size_t)(bm + row) * DIM + s_k4]);
        rb[i] = *reinterpret_cast<const float4*>(&X[(size_t)(bn + row) * DIM + s_k4]);
    }

    v8f acc[2][4] = {};

    for (int kc = 0; kc < DIM; kc += KT) {
        __syncthreads();                  // previous compute done with LDS
#pragma unroll
        for (int i = 0; i < 4; ++i) {
            int row = s_row + 32 * i;
            *reinterpret_cast<float4*>(&As[row][s_k4]) = ra[i];
            *reinterpret_cast<float4*>(&Bs[row][s_k4]) = rb[i];
        }
        __syncthreads();

        if (kc + KT < DIM) {              // prefetch next chunk during compute
#pragma unroll
            for (int i = 0; i < 4; ++i) {
                int row = s_row + 32 * i;
                ra[i] = *reinterpret_cast<const float4*>(
                    &X[(size_t)(bm + row) * DIM + kc + KT + s_k4]);
                rb[i] = *reinterpret_cast<const float4*>(
                    &X[(size_t)(bn + row) * DIM + kc + KT + s_k4]);
            }
        }

#pragma unroll
        for (int ks = 0; ks < KT; ks += 4) {
            // A frag (16x4 f32): lane half selects K pair -> adjacent b64
            v2f afrag[2], bfrag[4];
#pragma unroll
            for (int mi = 0; mi < 2; ++mi) {
                int m = waveM * 32 + mi * 16 + l16;
                afrag[mi] = *reinterpret_cast<const v2f*>(&As[m][ks + 2 * half]);
            }
#pragma unroll
            for (int ni = 0; ni < 4; ++ni) {
                int n = waveN * 64 + ni * 16 + l16;
                bfrag[ni] = *reinterpret_cast<const v2f*>(&Bs[n][ks + 2 * half]);
            }
#pragma unroll
            for (int mi = 0; mi < 2; ++mi)
#pragma unroll
                for (int ni = 0; ni < 4; ++ni)
                    acc[mi][ni] = __builtin_amdgcn_wmma_f32_16x16x4_f32(
                        false, afrag[mi], false, bfrag[ni],
                        (short)0, acc[mi][ni], false, false);
        }
    }

    // dist = sqrt(max(sq_i + sq_j - 2 G_ij, 1e-12)).
    // C/D layout: VGPR r -> row (r + 8*half), col (lane&15)  [ISA 7.12.2]
    // Pass 1: mine for tile rows (bm + rL), reduce over the 4 ni sub-tiles.
#pragma unroll
    for (int mi = 0; mi < 2; ++mi) {
#pragma unroll
        for (int r = 0; r < 8; ++r) {
            int   rL   = waveM * 32 + mi * 16 + r + 8 * half;
            float srow = sqA[rL];
            int   trow = tA[rL];
            unsigned long long bestAp = 0ull, bestAn = ~0ull;
#pragma unroll
            for (int ni = 0; ni < 4; ++ni) {
                int   cL   = waveN * 64 + ni * 16 + l16;
                float d2   = srow + sqB[cL] - 2.0f * acc[mi][ni][r];
                float dist = sqrtf(fmaxf(d2, 1e-12f));
                unsigned int bits = __float_as_uint(dist);  // dist > 0: monotonic
                unsigned int col  = (unsigned int)(bn + cL);
                if (trow == tB[cL]) {
                    unsigned long long p =
                        ((unsigned long long)bits << 32) | (unsigned int)(~col);
                    bestAp = (p > bestAp) ? p : bestAp;     // ties -> smaller col
                } else {
                    unsigned long long p =
                        ((unsigned long long)bits << 32) | col;
                    bestAn = (p < bestAn) ? p : bestAn;     // ties -> smaller col
                }
            }
            if (bestAp) atomicMax(&apL[rL], bestAp);
            if (bestAn != ~0ull) atomicMin(&anL[rL], bestAn);
        }
    }
    // Pass 2 (symmetry): same elements mine for rows (bn + cL); for fixed ni
    // all 16 (mi,r) elements of a lane share one target row -> local reduce.
#pragma unroll
    for (int ni = 0; ni < 4; ++ni) {
        int   cL   = waveN * 64 + ni * 16 + l16;
        float scol = sqB[cL];
        int   tcol = tB[cL];
        unsigned long long bestAp = 0ull, bestAn = ~0ull;
#pragma unroll
        for (int mi = 0; mi < 2; ++mi) {
#pragma unroll
            for (int r = 0; r < 8; ++r) {
                int   rL   = waveM * 32 + mi * 16 + r + 8 * half;
                float d2   = scol + sqA[rL] - 2.0f * acc[mi][ni][r];
                float dist = sqrtf(fmaxf(d2, 1e-12f));
                unsigned int bits = __float_as_uint(dist);
                unsigned int col  = (unsigned int)(bm + rL);
                if (tcol == tA[rL]) {
                    unsigned long long p =
                        ((unsigned long long)bits << 32) | (unsigned int)(~col);
                    bestAp = (p > bestAp) ? p : bestAp;
                } else {
                    unsigned long long p =
                        ((unsigned long long)bits << 32) | col;
                    bestAn = (p < bestAn) ? p : bestAn;
                }
            }
        }
        if (bestAp) atomicMax(&apLB[cL], bestAp);
        if (bestAn != ~0ull) atomicMin(&anLB[cL], bestAn);
    }
    __syncthreads();
    if (tid < TILE) {
        if (apL[tid]) atomicMax(&ap[bm + tid], apL[tid]);
        if (anL[tid] != ~0ull) atomicMin(&an[bm + tid], anL[tid]);
        if (apLB[tid]) atomicMax(&ap[bn + tid], apLB[tid]);
        if (anLB[tid] != ~0ull) atomicMin(&an[bn + tid], anLB[tid]);
    }
}

__global__ __launch_bounds__(256)
void mine_finish_kernel(const unsigned long long* __restrict__ ap,
                        const unsigned long long* __restrict__ an,
                        int* __restrict__ p_inds, int* __restrict__ n_inds,
                        float* __restrict__ sums) {
    int i = blockIdx.x * blockDim.x + threadIdx.x;
    float term = 0.0f;
    if (i < N_SAMP) {
        unsigned long long a = ap[i], b = an[i];
        float dap = __uint_as_float((unsigned int)(a >> 32));
        float dan = __uint_as_float((unsigned int)(b >> 32));
        p_inds[i] = (int)(~(unsigned int)a);
        n_inds[i] = (int)((unsigned int)b);
        term = fmaxf(dap - dan + MARGIN_F, 0.0f);
    }
    __shared__ float red[256];
    red[threadIdx.x] = term; __syncthreads();
    for (int w = 128; w > 0; w >>= 1) {
        if (threadIdx.x < w) red[threadIdx.x] += red[threadIdx.x + w];
        __syncthreads();
    }
    if (threadIdx.x == 0) atomicAdd(&sums[0], red[0]);
}

__device__ float shortest_path8(const float m[NLOC][NLOC]) {
    float r[NLOC];
    r[0] = m[0][0];
    for (int b = 1; b < NLOC; ++b) r[b] = r[b - 1] + m[0][b];
    for (int a = 1; a < NLOC; ++a) {
        r[0] += m[a][0];
        for (int b = 1; b < NLOC; ++b)
            r[b] = fminf(r[b], r[b - 1]) + m[a][b];
    }
    return r[NLOC - 1];
}

// One block (64 thr) per sample: 8x8 local distance matrices vs hard pos/neg,
// squash (exp(d)-1)/(exp(d)+1), 8x8 shortest-path DP.
__global__ __launch_bounds__(64)
void local_kernel(const float* __restrict__ LF,
                  const int* __restrict__ p_inds,
                  const int* __restrict__ n_inds,
                  float* __restrict__ sums) {
    __shared__ float Si[NLOC * DLOC], Sp[NLOC * DLOC], Sn[NLOC * DLOC];
    __shared__ float dp[NLOC][NLOC], dn[NLOC][NLOC];
    __shared__ float xx[NLOC], pp[NLOC], nn[NLOC];
    const int i = blockIdx.x;
    const int t = threadIdx.x;
    const int p = p_inds[i];
    const int n = n_inds[i];
    // LF layout [4096][128][8]; lf[a][f] = LF[i*1024 + f*8 + a]
#pragma unroll
    for (int j = 0; j < 4; ++j) {
        int off = (t + 64 * j) * 4;
        *reinterpret_cast<float4*>(&Si[off]) =
            *reinterpret_cast<const float4*>(&LF[(size_t)i * 1024 + off]);
        *reinterpret_cast<float4*>(&Sp[off]) =
            *reinterpret_cast<const float4*>(&LF[(size_t)p * 1024 + off]);
        *reinterpret_cast<float4*>(&Sn[off]) =
            *reinterpret_cast<const float4*>(&LF[(size_t)n * 1024 + off]);
    }
    __syncthreads();
    if (t < 24) {
        const float* S = (t < 8) ? Si : ((t < 16) ? Sp : Sn);
        int a = t & 7;
        float s = 0.0f;
        for (int f = 0; f < DLOC; ++f) { float v = S[f * 8 + a]; s += v * v; }
        if (t < 8) xx[a] = s; else if (t < 16) pp[a] = s; else nn[a] = s;
    }
    __syncthreads();
    {
        int a = t >> 3, b = t & 7;
        float dotp = 0.0f, dotn = 0.0f;
        for (int f = 0; f < DLOC; ++f) {
            float v = Si[f * 8 + a];
            dotp += v * Sp[f * 8 + b];
            dotn += v * Sn[f * 8 + b];
        }
        float ddp = sqrtf(fmaxf(xx[a] + pp[b] - 2.0f * dotp, 1e-12f));
        float ddn = sqrtf(fmaxf(xx[a] + nn[b] - 2.0f * dotn, 1e-12f));
        float ep = expf(ddp);
        float en = expf(ddn);
        dp[a][b] = (ep - 1.0f) / (ep + 1.0f);
        dn[a][b] = (en - 1.0f) / (en + 1.0f);
    }
    __syncthreads();
    if (t == 0) {
        float lap = shortest_path8(dp);
        float lan = shortest_path8(dn);
        atomicAdd(&sums[1], fmaxf(lap - lan + MARGIN_F, 0.0f));
    }
}

__global__ void finalize_kernel(const float* __restrict__ sums,
                                float* __restrict__ out) {
    if (threadIdx.x == 0) {
        out[0] = sums[0] * (1.0f / N_SAMP);
        out[1] = sums[1] * (1.0f / N_SAMP);
    }
}

extern "C" void kernel_launch(void* const* d_in, const int* in_sizes, int n_in,
                              void* d_out, int out_size, void* d_ws, size_t ws_size,
                              hipStream_t stream) {
    const float* X       = (const float*)d_in[0];   // [4096, 2048] f32
    const int*   targets = (const int*)d_in[1];     // [4096] int
    const float* LF      = (const float*)d_in[2];   // [4096, 128, 8] f32

    char* ws = (char*)d_ws;
    float*              sq     = (float*)(ws);
    unsigned long long* ap     = (unsigned long long*)(ws + 16384);
    unsigned long long* an     = (unsigned long long*)(ws + 49152);
    int*                p_inds = (int*)(ws + 81920);
    int*                n_inds = (int*)(ws + 98304);
    float*              sums   = (float*)(ws + 114688);
    float*              out    = (float*)d_out;

    init_kernel<<<16, 256, 0, stream>>>(ap, an, sums);
    sq_kernel<<<N_SAMP, 256, 0, stream>>>(X, sq);
    const int ntri = NTILE * (NTILE + 1) / 2;       // 528 lower-triangle tiles
    gemm_mine_kernel<<<ntri, 256, 0, stream>>>(X, targets, sq, ap, an);
    mine_finish_kernel<<<N_SAMP / 256, 256, 0, stream>>>(ap, an, p_inds, n_inds, sums);
    local_kernel<<<N_SAMP, 64, 0, stream>>>(LF, p_inds, n_inds, sums);
    finalize_kernel<<<1, 32, 0, stream>>>(sums, out);
}